// Invo2D_70738111365745
// MI455X (gfx1250) — compile-verified
//
#include <hip/hip_runtime.h>
#include <hip/hip_bf16.h>

// Invo2D for MI455X (gfx1250, wave32) — fully fused single kernel.
//
// Shapes: B=8, H=64, W=64, C=256, red=64, K=3, G=16, GC=16, kk*G=144.
// Memory-bound: ~1.8 GFLOP vs ~100 MB traffic. Fusion keeps the two
// intermediate tensors (T: 16 MB roundtrip, WGT: 38 MB roundtrip) entirely
// in LDS, cutting HBM traffic ~36% vs the 3-kernel version.
//
// One wave handles one 16-pixel M-tile (2048 tiles). Each wave uses a
// private LDS slab -> wave-synchronous, no barriers. GEMMs use exact fp32
// math via V_WMMA_F32_16X16X4_F32.
//
// Fragment layouts (ISA 7.12.2, wave32):
//   A 16x4 f32: lanes 0-15 -> M=lane, VGPR0=K0, VGPR1=K1;
//               lanes 16-31 -> M=lane-16, VGPR0=K2, VGPR1=K3.
//   B 4x16 f32 (mirrored): lane&15 = N; rows K0/K1 (lanes<16) or K2/K3.
//   C/D 16x16 f32: 8 VGPRs, element (M = v + 8*(lane>>4), N = lane&15).

typedef float v2f __attribute__((ext_vector_type(2)));
typedef float v8f __attribute__((ext_vector_type(8)));

#define C_IN    256
#define RED     64
#define N2      144
// LDS row strides (floats), padded for bank-conflict-free access:
//  TSTRIDE=68: A-fragment ds_load_b64 -> banks 4r+c+{0..3} span all 64 banks.
//  WSTRIDE=148: half-wave row offset 8*148 mod 64 = 32 -> disjoint bank sets.
#define TSTRIDE 68
#define WSTRIDE 148
#define WAVES_PER_BLOCK 4
#define SLAB (16 * TSTRIDE + 16 * WSTRIDE)   // floats per wave

__global__ __launch_bounds__(128) void invo2d_fused_kernel(
    const float* __restrict__ X, const float* __restrict__ W1,
    const float* __restrict__ b1, const float* __restrict__ W2,
    const float* __restrict__ b2, float* __restrict__ OUT) {
  __shared__ float smem[WAVES_PER_BLOCK * SLAB];   // 55296 B

  const int lane = threadIdx.x & 31;
  const int wave = threadIdx.x >> 5;
  float* Tb = smem + wave * SLAB;       // [16][TSTRIDE] : T tile (16x64)
  float* Wb = Tb + 16 * TSTRIDE;        // [16][WSTRIDE] : WGT tile (16x144)

  const int tile = blockIdx.x * WAVES_PER_BLOCK + wave;  // 0..2047
  const int r0   = tile * 16;           // first pixel (row) of this tile
  const int mrow = lane & 15;
  const int half = lane >> 4;           // 0: K0/K1, 1: K2/K3
  const int kOff = half << 1;
  const int arow = r0 + mrow;

  // ---------------- Phase 1: T = X @ W1 + b1  (16x64) ----------------
  v8f acc1[4] = {};
  for (int k0 = 0; k0 < C_IN; k0 += 4) {
    const v2f a = *(const v2f*)(X + (size_t)arow * C_IN + (k0 + kOff));
#pragma unroll
    for (int ct = 0; ct < 4; ++ct) {
      v2f b;
      b.x = W1[(size_t)(k0 + kOff) * RED + ct * 16 + mrow];
      b.y = W1[(size_t)(k0 + kOff + 1) * RED + ct * 16 + mrow];
      acc1[ct] = __builtin_amdgcn_wmma_f32_16x16x4_f32(
          false, a, false, b, (short)0, acc1[ct], false, false);
    }
  }
#pragma unroll
  for (int ct = 0; ct < 4; ++ct) {
    const int col = ct * 16 + mrow;
    const float bias = b1[col];
#pragma unroll
    for (int v = 0; v < 8; ++v)
      Tb[(v + half * 8) * TSTRIDE + col] = acc1[ct][v] + bias;
  }

  // ---------------- Phase 2: WGT = T @ W2 + b2  (16x144) ----------------
  // Wave-synchronous: same wave wrote Tb; LDS ops are in-order per wave.
  v8f acc2[9] = {};
  for (int k0 = 0; k0 < RED; k0 += 4) {
    const v2f a = *(const v2f*)(Tb + mrow * TSTRIDE + (k0 + kOff));
#pragma unroll
    for (int ct = 0; ct < 9; ++ct) {
      v2f b;
      b.x = W2[(size_t)(k0 + kOff) * N2 + ct * 16 + mrow];
      b.y = W2[(size_t)(k0 + kOff + 1) * N2 + ct * 16 + mrow];
      acc2[ct] = __builtin_amdgcn_wmma_f32_16x16x4_f32(
          false, a, false, b, (short)0, acc2[ct], false, false);
    }
  }
#pragma unroll
  for (int ct = 0; ct < 9; ++ct) {
    const int col = ct * 16 + mrow;
    const float bias = b2[col];
#pragma unroll
    for (int v = 0; v < 8; ++v)
      Wb[(v + half * 8) * WSTRIDE + col] = acc2[ct][v] + bias;
  }

  // ---------------- Phase 3: involution gather ----------------
  // out[pix, ch] = sum_k Wb[p][g*9+k] * x[b, h+di, w+dj, csrc]
  //   g=ch/16, gc=ch%16, f = g*144 + gc*9 + k, kpos=f/256, csrc=f%255+...=f&255
  for (int p = 0; p < 16; ++p) {
    const int pix = r0 + p;
    const int w = pix & 63;
    const int h = (pix >> 6) & 63;
    const int b = pix >> 12;
    const float* wrow = Wb + p * WSTRIDE;
    for (int j = 0; j < 8; ++j) {
      const int ch = j * 32 + lane;       // each ch exactly once per wave
      const int g  = ch >> 4;
      const int gc = ch & 15;
      const int fbase = g * 144 + gc * 9;
      const float* wp = wrow + g * 9;     // broadcast LDS reads
      float acc = 0.0f;
#pragma unroll
      for (int k = 0; k < 9; ++k) {
        const int f    = fbase + k;       // < 2304
        const int kpos = f >> 8;          // 0..8
        const int csrc = f & 255;
        const int hh = h + kpos / 3 - 1;
        const int ww = w + kpos % 3 - 1;
        float xv = 0.0f;
        if ((unsigned)hh < 64u && (unsigned)ww < 64u)
          xv = X[((((size_t)b * 64 + hh) * 64) + ww) * 256 + csrc];
        acc = fmaf(wp[k], xv, acc);
      }
      OUT[(size_t)pix * 256 + ch] = acc;  // coalesced b32 stores
    }
  }
}

// ---------------------------------------------------------------------------
extern "C" void kernel_launch(void* const* d_in, const int* in_sizes, int n_in,
                              void* d_out, int out_size, void* d_ws, size_t ws_size,
                              hipStream_t stream) {
  const float* x  = (const float*)d_in[0];   // [8,64,64,256]
  const float* W1 = (const float*)d_in[1];   // [256,64]
  const float* b1 = (const float*)d_in[2];   // [64]
  const float* W2 = (const float*)d_in[3];   // [64,144]
  const float* b2 = (const float*)d_in[4];   // [144]
  float* out = (float*)d_out;                // [8,64,64,256]

  // 2048 row-tiles, 4 waves (tiles) per 128-thread block -> 512 blocks.
  invo2d_fused_kernel<<<512, 128, 0, stream>>>(x, W1, b1, W2, b2, out);
}